// Vista2DLoss_68238440399209
// MI455X (gfx1250) — compile-verified
//
#include <hip/hip_runtime.h>
#include <hip/hip_bf16.h>
#include <math.h>

typedef __attribute__((ext_vector_type(2))) float v2f;
typedef __attribute__((ext_vector_type(8))) float v8f;

// ---------------- problem constants ----------------
constexpr int Bn   = 8;
constexpr int Hh   = 384;
constexpr int Ww   = 384;
constexpr int P    = Hh * Ww;        // 147456 pixels / image
constexpr int K    = 33;             // instance ids (0 = background)
constexpr int E    = 8;              // embedding dim
constexpr int TPB  = 256;            // threads per block (8 waves on wave32)
constexpr int PIXPB = 4096;          // pixels per block
constexpr int CHUNKS = P / PIXPB;    // 36 blocks per image
constexpr int NB1  = Bn * CHUNKS;    // 288 total blocks in per-pixel passes
constexpr float W_EDGE = 10.0f;
constexpr float DELTA_V = 0.5f;
constexpr float DELTA_D = 1.5f;

// ---------------- workspace layout (floats) ----------------
constexpr size_t OFF_P1    = 0;                      // [NB1][330] segment partials
constexpr size_t OFF_P1S   = OFF_P1   + (size_t)NB1 * 330;  // [NB1][4] scalar partials
constexpr size_t OFF_CNT   = OFF_P1S  + (size_t)NB1 * 4;    // [264]
constexpr size_t OFF_CENT  = OFF_CNT  + 264;                // [264][8]
constexpr size_t OFF_PRES  = OFF_CENT + 264 * 8;            // [264]
constexpr size_t OFF_PUSHB = OFF_PRES + 264;                // [8]
constexpr size_t OFF_NORMB = OFF_PUSHB + 8;                 // [8]
constexpr size_t OFF_NPRES = OFF_NORMB + 8;                 // [8]
constexpr size_t OFF_SCAL  = OFF_NPRES + 8;                 // [4] sem, adiff, agrid, argba
constexpr size_t OFF_P3    = OFF_SCAL  + 4;                 // [NB1][33] pull partials

// =====================================================================
// Kernel 1: per-pixel pass. Computes per-block partials for:
//   - semantic NLL sum, three L1 sums
//   - per-(b,k): cnt, sum(w), sum(w*emb[e])  (boundary-weighted)
// Uses the fact labels are 16x16-blocky: each 16-lane half of a wave32 is
// usually label-uniform -> half-wave shfl reduction, 1 LDS atomic / half.
// =====================================================================
__global__ __launch_bounds__(TPB) void k1_main(
    const float* __restrict__ sem, const int* __restrict__ cls,
    const float* __restrict__ inst, const float* __restrict__ geo,
    const float* __restrict__ gdiff, const float* __restrict__ ggrid,
    const float* __restrict__ grgba, const int* __restrict__ labels,
    float* __restrict__ ws)
{
    __shared__ float lseg[K * 10];   // [k][0]=cnt [1]=wsum [2..9]=ew
    __shared__ float lscal[8 * 4];   // per-wave scalar partials
    const int b = blockIdx.y, chunk = blockIdx.x;
    const int tid = threadIdx.x, lane = tid & 31;

    for (int i = tid; i < K * 10; i += TPB) lseg[i] = 0.0f;
    __syncthreads();

    const int* lab = labels + (size_t)b * P;
    const int base = chunk * PIXPB;
    float s_sem = 0.f, s_d = 0.f, s_g = 0.f, s_r = 0.f;

    for (int it = 0; it < PIXPB / TPB; ++it) {
        const int p = base + it * TPB + tid;
        const int y = p / Ww, x = p - y * Ww;
        const int l = lab[p];

        // boundary weight: 3x3 min/max with replicate padding
        int mn = l, mx = l;
        const int y0 = (y > 0) ? y - 1 : 0, y1 = (y < Hh - 1) ? y + 1 : y;
        const int x0 = (x > 0) ? x - 1 : 0, x1 = (x < Ww - 1) ? x + 1 : x;
        for (int yy = y0; yy <= y1; ++yy)
            for (int xx = x0; xx <= x1; ++xx) {
                const int v = lab[yy * Ww + xx];
                mn = min(mn, v); mx = max(mx, v);
            }
        const float w = (mx != mn) ? W_EDGE : 1.0f;

        // semantic NLL (log-softmax over 4 channels)
        {
            const float v0 = sem[((size_t)b * 4 + 0) * P + p];
            const float v1 = sem[((size_t)b * 4 + 1) * P + p];
            const float v2 = sem[((size_t)b * 4 + 2) * P + p];
            const float v3 = sem[((size_t)b * 4 + 3) * P + p];
            const float m  = fmaxf(fmaxf(v0, v1), fmaxf(v2, v3));
            const float s  = expf(v0 - m) + expf(v1 - m) + expf(v2 - m) + expf(v3 - m);
            const float lse = m + logf(s);
            const int c = cls[(size_t)b * P + p];
            const float vc = (c == 0) ? v0 : (c == 1) ? v1 : (c == 2) ? v2 : v3;
            s_sem += lse - vc;
        }

        // geometry L1 terms
        const float* gb = geo   + (size_t)b * 16 * P + p;
        const float* dd = gdiff + (size_t)b * 9  * P + p;
        #pragma unroll
        for (int c = 0; c < 9; ++c) s_d += fabsf(gb[(size_t)c * P] - dd[(size_t)c * P]);
        const float* gg = ggrid + (size_t)b * 3 * P + p;
        #pragma unroll
        for (int c = 0; c < 3; ++c) s_g += fabsf(gb[(size_t)(9 + c) * P] - gg[(size_t)c * P]);
        const float* gr = grgba + (size_t)b * 4 * P + p;
        #pragma unroll
        for (int c = 0; c < 4; ++c) s_r += fabsf(gb[(size_t)(12 + c) * P] - gr[(size_t)c * P]);

        // instance segment accumulation
        float vals[10];
        vals[0] = 1.0f; vals[1] = w;
        const float* eb = inst + (size_t)b * 8 * P + p;
        #pragma unroll
        for (int e = 0; e < 8; ++e) vals[2 + e] = eb[(size_t)e * P] * w;

        const int llead = __shfl(l, lane & 16);          // leader of this 16-lane half
        const bool halfuni = (__ballot(l == llead) == 0xFFFFFFFFull);
        if (halfuni) {
            #pragma unroll
            for (int j = 0; j < 10; ++j) {
                float v = vals[j];
                v += __shfl_down(v, 8, 16);
                v += __shfl_down(v, 4, 16);
                v += __shfl_down(v, 2, 16);
                v += __shfl_down(v, 1, 16);
                vals[j] = v;
            }
            if ((lane & 15) == 0)
                #pragma unroll
                for (int j = 0; j < 10; ++j) atomicAdd(&lseg[l * 10 + j], vals[j]);
        } else {
            #pragma unroll
            for (int j = 0; j < 10; ++j) atomicAdd(&lseg[l * 10 + j], vals[j]);
        }
    }

    // wave-reduce scalars, then block-combine
    #pragma unroll
    for (int off = 16; off > 0; off >>= 1) {
        s_sem += __shfl_down(s_sem, off);
        s_d   += __shfl_down(s_d, off);
        s_g   += __shfl_down(s_g, off);
        s_r   += __shfl_down(s_r, off);
    }
    const int wv = tid >> 5;
    if (lane == 0) {
        lscal[wv * 4 + 0] = s_sem; lscal[wv * 4 + 1] = s_d;
        lscal[wv * 4 + 2] = s_g;   lscal[wv * 4 + 3] = s_r;
    }
    __syncthreads();

    const int blk = b * CHUNKS + chunk;
    float* p1 = ws + OFF_P1 + (size_t)blk * 330;
    for (int i = tid; i < K * 10; i += TPB) p1[i] = lseg[i];
    if (tid == 0) {
        float a = 0.f, bb = 0.f, c = 0.f, d = 0.f;
        for (int w2 = 0; w2 < 8; ++w2) {
            a += lscal[w2 * 4 + 0]; bb += lscal[w2 * 4 + 1];
            c += lscal[w2 * 4 + 2]; d  += lscal[w2 * 4 + 3];
        }
        float* ps = ws + OFF_P1S + (size_t)blk * 4;
        ps[0] = a; ps[1] = bb; ps[2] = c; ps[3] = d;
    }
}

// =====================================================================
// Kernel 2 (1 block, 256 threads = 8 waves): deterministic reduction of
// block partials -> centers/presence/norm; then each wave computes its
// batch's center Gram matrix with V_WMMA_F32_16X16X4_F32 (full fp32) and
// derives the push loss from pd^2 = |ci|^2 + |cj|^2 - 2*Gij.
// =====================================================================
__global__ __launch_bounds__(TPB) void k2_centers_push(float* __restrict__ ws)
{
    __shared__ float lc[Bn * 48 * E];   // centers, rows padded 33->48 with zeros
    __shared__ float ln[Bn * 48];       // |c|^2
    __shared__ float lpres[Bn * 48];
    __shared__ float lnpres[Bn];
    const int tid = threadIdx.x, lane = tid & 31;

    for (int i = tid; i < Bn * 48 * E; i += TPB) lc[i] = 0.0f;
    for (int i = tid; i < Bn * 48; i += TPB) { ln[i] = 0.0f; lpres[i] = 0.0f; }
    __syncthreads();

    for (int seg = tid; seg < Bn * K; seg += TPB) {
        const int b = seg / K, k = seg - b * K;
        float s[10];
        #pragma unroll
        for (int j = 0; j < 10; ++j) s[j] = 0.0f;
        for (int blk = 0; blk < CHUNKS; ++blk) {
            const float* p1 = ws + OFF_P1 + (size_t)(b * CHUNKS + blk) * 330 + k * 10;
            #pragma unroll
            for (int j = 0; j < 10; ++j) s[j] += p1[j];
        }
        const float cnt = s[0], wsum = s[1];
        ws[OFF_CNT + seg] = cnt;
        const float pres = (cnt > 0.0f && k > 0) ? 1.0f : 0.0f;
        ws[OFF_PRES + seg] = pres;
        lpres[b * 48 + k] = pres;
        float n2 = 0.0f;
        #pragma unroll
        for (int e = 0; e < E; ++e) {
            const float c = s[2 + e] / (wsum + 1e-8f);
            ws[OFF_CENT + (size_t)seg * E + e] = c;
            lc[(b * 48 + k) * E + e] = c;
            n2 += c * c;
        }
        ln[b * 48 + k] = n2;
    }
    if (tid < 4) {   // global scalar sums, fixed order (deterministic)
        double acc = 0.0;
        for (int blk = 0; blk < NB1; ++blk) acc += (double)ws[OFF_P1S + (size_t)blk * 4 + tid];
        ws[OFF_SCAL + tid] = (float)acc;
    }
    __syncthreads();

    if (tid < Bn) {  // per-batch npres + norm loss, serial (deterministic)
        const int b = tid;
        double nrm = 0.0; int np = 0;
        for (int k = 1; k < K; ++k)
            if (lpres[b * 48 + k] > 0.0f) {
                ++np;
                nrm += sqrt(fmax((double)ln[b * 48 + k], 1e-12));
            }
        lnpres[b] = (float)np;
        ws[OFF_NPRES + b] = (float)np;
        ws[OFF_NORMB + b] = (float)((np > 0) ? nrm / np : 0.0);
    }
    __syncthreads();

    // ---- WMMA Gram: wave w handles batch w (8 waves, EXEC all ones) ----
    const int b    = tid >> 5;
    const int half = lane >> 4;
    const int m    = lane & 15;
    float pushacc = 0.0f;
    for (int ti = 0; ti < 3; ++ti) {
        for (int tj = ti; tj < 3; ++tj) {
            // fp32 16x16x4 fragments: VGPR0 = K(2h), VGPR1 = K(2h+1);
            // A(16x4) and B(4x16)=A^T share the identical per-lane layout.
            const int ra = ti * 16 + m, rb = tj * 16 + m;
            v2f a0, a1, b0, b1;
            a0.x = lc[(b * 48 + ra) * E + 2 * half + 0];
            a0.y = lc[(b * 48 + ra) * E + 2 * half + 1];
            a1.x = lc[(b * 48 + ra) * E + 4 + 2 * half + 0];
            a1.y = lc[(b * 48 + ra) * E + 4 + 2 * half + 1];
            b0.x = lc[(b * 48 + rb) * E + 2 * half + 0];
            b0.y = lc[(b * 48 + rb) * E + 2 * half + 1];
            b1.x = lc[(b * 48 + rb) * E + 4 + 2 * half + 0];
            b1.y = lc[(b * 48 + rb) * E + 4 + 2 * half + 1];
            v8f acc = {};
            acc = __builtin_amdgcn_wmma_f32_16x16x4_f32(false, a0, false, b0,
                                                        (short)0, acc, false, false);
            acc = __builtin_amdgcn_wmma_f32_16x16x4_f32(false, a1, false, b1,
                                                        (short)0, acc, false, false);
            #pragma unroll
            for (int r = 0; r < 8; ++r) {
                const int gi = ti * 16 + r + 8 * half;   // D row (C/D VGPR layout)
                const int gj = tj * 16 + m;              // D col
                if (gi >= 1 && gj >= 1 && gi < K && gj < K && gi < gj &&
                    lpres[b * 48 + gi] > 0.0f && lpres[b * 48 + gj] > 0.0f) {
                    const float pd2 = ln[b * 48 + gi] + ln[b * 48 + gj] - 2.0f * acc[r];
                    const float pd  = sqrtf(fmaxf(pd2, 1e-12f));
                    const float h   = 2.0f * DELTA_D - pd;
                    if (h > 0.0f) pushacc += h * h;
                }
            }
        }
    }
    #pragma unroll
    for (int off = 16; off > 0; off >>= 1) pushacc += __shfl_down(pushacc, off);
    if (lane == 0) {
        const float np = lnpres[b];
        const float npairs = np * (np - 1.0f) * 0.5f;
        ws[OFF_PUSHB + b] = pushacc / fmaxf(npairs, 1.0f);
    }
}

// =====================================================================
// Kernel 3: per-pixel pull pass against centers.
// =====================================================================
__global__ __launch_bounds__(TPB) void k3_pull(
    const float* __restrict__ inst, const int* __restrict__ labels,
    float* __restrict__ ws)
{
    __shared__ float lcent[K * E];
    __shared__ float lpull[K];
    const int b = blockIdx.y, chunk = blockIdx.x;
    const int tid = threadIdx.x, lane = tid & 31;

    for (int i = tid; i < K * E; i += TPB) lcent[i] = ws[OFF_CENT + (size_t)b * K * E + i];
    for (int i = tid; i < K; i += TPB) lpull[i] = 0.0f;
    __syncthreads();

    const int* lab = labels + (size_t)b * P;
    const float* eb0 = inst + (size_t)b * 8 * P;
    const int base = chunk * PIXPB;

    for (int it = 0; it < PIXPB / TPB; ++it) {
        const int p = base + it * TPB + tid;
        const int y = p / Ww, x = p - y * Ww;
        const int l = lab[p];

        int mn = l, mx = l;
        const int y0 = (y > 0) ? y - 1 : 0, y1 = (y < Hh - 1) ? y + 1 : y;
        const int x0 = (x > 0) ? x - 1 : 0, x1 = (x < Ww - 1) ? x + 1 : x;
        for (int yy = y0; yy <= y1; ++yy)
            for (int xx = x0; xx <= x1; ++xx) {
                const int v = lab[yy * Ww + xx];
                mn = min(mn, v); mx = max(mx, v);
            }
        const float w = (mx != mn) ? W_EDGE : 1.0f;

        float d2 = 0.0f;
        #pragma unroll
        for (int e = 0; e < E; ++e) {
            const float df = eb0[(size_t)e * P + p] - lcent[l * E + e];
            d2 += df * df;
        }
        const float dist = sqrtf(fmaxf(d2, 1e-12f));
        const float t = dist - DELTA_V;
        float pull = (t > 0.0f ? t * t : 0.0f) * w;
        if (l == 0) pull = 0.0f;   // background masked out downstream anyway

        const int llead = __shfl(l, lane & 16);
        const bool halfuni = (__ballot(l == llead) == 0xFFFFFFFFull);
        if (halfuni) {
            float v = pull;
            v += __shfl_down(v, 8, 16);
            v += __shfl_down(v, 4, 16);
            v += __shfl_down(v, 2, 16);
            v += __shfl_down(v, 1, 16);
            if ((lane & 15) == 0) atomicAdd(&lpull[l], v);
        } else if (l > 0) {
            atomicAdd(&lpull[l], pull);
        }
    }
    __syncthreads();
    float* p3 = ws + OFF_P3 + (size_t)(b * CHUNKS + chunk) * K;
    for (int i = tid; i < K; i += TPB) p3[i] = lpull[i];
}

// =====================================================================
// Kernel 4 (1 block): deterministic final combine -> 4 scalar outputs.
// =====================================================================
__global__ __launch_bounds__(TPB) void k4_final(const float* __restrict__ ws,
                                                float* __restrict__ out)
{
    __shared__ float lsegv[Bn * K];
    const int tid = threadIdx.x;
    for (int seg = tid; seg < Bn * K; seg += TPB) {
        const int b = seg / K, k = seg - b * K;
        float s = 0.0f;
        for (int blk = 0; blk < CHUNKS; ++blk)
            s += ws[OFF_P3 + (size_t)(b * CHUNKS + blk) * K + k];
        const float cnt  = ws[OFF_CNT + seg];
        const float pres = ws[OFF_PRES + seg];
        lsegv[seg] = pres * (s / fmaxf(cnt, 1.0f));
    }
    __syncthreads();
    if (tid == 0) {
        double lpull = 0.0;
        for (int i = 0; i < Bn * K; ++i) lpull += (double)lsegv[i];
        double lpush = 0.0, lnorm = 0.0, nb = 0.0;
        for (int b = 0; b < Bn; ++b) {
            lpush += (double)ws[OFF_PUSHB + b];
            lnorm += (double)ws[OFF_NORMB + b];
            if (ws[OFF_NPRES + b] > 0.0f) nb += 1.0;
        }
        const double n = fmax(nb, 1.0);
        const double lins = (lpull + lpush + 0.001 * lnorm) / n;
        const double lsem = (double)ws[OFF_SCAL + 0] / ((double)Bn * P);
        const double laff = (double)ws[OFF_SCAL + 1] / ((double)Bn * 9 * P)
                          + (double)ws[OFF_SCAL + 2] / ((double)Bn * 3 * P)
                          + (double)ws[OFF_SCAL + 3] / ((double)Bn * 4 * P);
        const double tot = lsem + laff + lins;
        out[0] = (float)tot;  out[1] = (float)lsem;
        out[2] = (float)laff; out[3] = (float)lins;
    }
}

extern "C" void kernel_launch(void* const* d_in, const int* in_sizes, int n_in,
                              void* d_out, int out_size, void* d_ws, size_t ws_size,
                              hipStream_t stream)
{
    const float* semantic     = (const float*)d_in[0];
    const int*   class_labels = (const int*)  d_in[1];
    const float* instance     = (const float*)d_in[2];
    const float* geometry     = (const float*)d_in[3];
    const float* gt_diff      = (const float*)d_in[4];
    const float* gt_grid      = (const float*)d_in[5];
    const float* gt_rgba      = (const float*)d_in[6];
    const int*   labels       = (const int*)  d_in[7];
    float* out = (float*)d_out;
    float* ws  = (float*)d_ws;

    dim3 grid(CHUNKS, Bn);
    k1_main<<<grid, TPB, 0, stream>>>(semantic, class_labels, instance, geometry,
                                      gt_diff, gt_grid, gt_rgba, labels, ws);
    k2_centers_push<<<1, TPB, 0, stream>>>(ws);
    k3_pull<<<grid, TPB, 0, stream>>>(instance, labels, ws);
    k4_final<<<1, TPB, 0, stream>>>(ws, out);
    (void)in_sizes; (void)n_in; (void)out_size; (void)ws_size;
}